// DNC_42039139893402
// MI455X (gfx1250) — compile-verified
//
#include <hip/hip_runtime.h>
#include <math.h>

// ---------------- problem constants ----------------
#define BATCH   16
#define TSTEPS  32
#define INDIM   256
#define HID     512
#define RH      4
#define CELLW   64
#define MEMN    512
#define IFACE   471
#define IFP     480      // padded xi width
#define G4H     2048
#define CATW    768      // [out(512) | read_vecs(256)]
#define NWG     16
#define NTHR    512
#define DNC_DELTA 1e-6f

typedef float v2f __attribute__((ext_vector_type(2)));
typedef float v8f __attribute__((ext_vector_type(8)));

#if __has_builtin(__builtin_amdgcn_wmma_f32_16x16x4_f32)
#define HAVE_WMMA_F32 1
#endif

// ---------------- workspace layout (floats) ----------------
// first 64 floats reserved for the grid barrier (cnt, gen)
#define OFF_H     ((size_t)64)
#define OFF_C     (OFF_H     + (size_t)BATCH*HID)
#define OFF_GATES (OFF_C     + (size_t)BATCH*HID)
#define OFF_XI    (OFF_GATES + (size_t)BATCH*G4H)
#define OFF_CAT   (OFF_XI    + (size_t)BATCH*IFP)
#define OFF_USAGE (OFF_CAT   + (size_t)BATCH*CATW)
#define OFF_PREC  (OFF_USAGE + (size_t)BATCH*MEMN)
#define OFF_WW    (OFF_PREC  + (size_t)BATCH*MEMN)
#define OFF_RW    (OFF_WW    + (size_t)BATCH*MEMN)
#define OFF_MEM   (OFF_RW    + (size_t)BATCH*RH*MEMN)
#define OFF_LINK  (OFF_MEM   + (size_t)BATCH*MEMN*CELLW)
#define WS_TOTALF (OFF_LINK  + (size_t)BATCH*MEMN*MEMN)

// ---------------- shared memory layout (floats) ----------------
// GEMM stages alias sm[0..4095] (16 waves x 256 tile-reduction buffer).
#define S_IFACE 0
#define S_RW    512
#define S_WW    2560
#define S_PREC  3072
#define S_WCW   3584
#define S_ALLOC 4096
#define S_FWD   4608
#define S_BWD   6656
#define S_CW    8704
#define S_SORT  10752
#define S_SCAN  11264
#define S_RED   11776
#define S_MISC  12288
#define SM_FLOATS 12320
#define MISC_NORM  0   // 5 key norms (4 read + 1 write)
#define MISC_MODES 5   // 12 read-mode softmax values
#define MISC_SUMWW 17

__device__ __forceinline__ float sigmoidf_(float x) { return 1.0f / (1.0f + expf(-x)); }
__device__ __forceinline__ float softplusf_(float x) { return (x > 20.0f) ? x : log1pf(expf(x)); }

// fp32 WMMA K-loop: one 16x16 tile, A rows = batch, B cols = weight rows.
// aP/bP are per-lane pointers already offset by koff = 2*(lane>=16).
__device__ __forceinline__ v8f wmma_loop(const float* __restrict__ aP,
                                         const float* __restrict__ bP,
                                         int kn, v8f acc) {
#ifdef HAVE_WMMA_F32
  for (int k = 0; k < kn; k += 4) {
    v2f av = *reinterpret_cast<const v2f*>(aP + k);
    v2f bv = *reinterpret_cast<const v2f*>(bP + k);
    acc = __builtin_amdgcn_wmma_f32_16x16x4_f32(false, av, false, bv,
                                                (short)0, acc, false, false);
  }
#else
  // placeholder so the file still compiles if the builtin is absent
  for (int k = 0; k < kn; k += 4) {
    v2f av = *reinterpret_cast<const v2f*>(aP + k);
    v2f bv = *reinterpret_cast<const v2f*>(bP + k);
    acc[0] += av.x * bv.x + av.y * bv.y;
  }
#endif
  return acc;
}

// sense-reversing grid barrier; cnt/gen live in ws (zeroed every launch)
__device__ __forceinline__ void grid_sync(unsigned* cnt, unsigned* gen) {
  __threadfence();
  __syncthreads();
  if (threadIdx.x == 0) {
    unsigned g = atomicAdd(gen, 0u);
    unsigned a = atomicAdd(cnt, 1u);
    if (a == NWG - 1u) {
      atomicExch(cnt, 0u);
      __threadfence();
      atomicAdd(gen, 1u);
    } else {
      while (atomicAdd(gen, 0u) == g) { __builtin_amdgcn_s_sleep(2); }
    }
  }
  __syncthreads();
  __threadfence();
}

__global__ __launch_bounds__(NTHR, 1)
void dnc_persistent(const float* __restrict__ X,
                    const float* __restrict__ Wih, const float* __restrict__ bih,
                    const float* __restrict__ Whh, const float* __restrict__ bhh,
                    const float* __restrict__ Wout, const float* __restrict__ bout,
                    const float* __restrict__ Wint, const float* __restrict__ bint,
                    const float* __restrict__ Wmem, const float* __restrict__ bmem,
                    float* __restrict__ OUTP, float* __restrict__ ws) {
  __shared__ float sm[SM_FLOATS];
  __shared__ int   sIdx[MEMN];

  unsigned* cnt = reinterpret_cast<unsigned*>(ws);
  unsigned* gen = cnt + 1;

  float* Hs    = ws + OFF_H;
  float* Cs    = ws + OFF_C;
  float* GATES = ws + OFF_GATES;
  float* XI    = ws + OFF_XI;
  float* CAT   = ws + OFF_CAT;
  float* USAGE = ws + OFF_USAGE;
  float* PRECB = ws + OFF_PREC;
  float* WW    = ws + OFF_WW;
  float* RW    = ws + OFF_RW;
  float* MEMB  = ws + OFF_MEM;
  float* LINKB = ws + OFF_LINK;

  const int tid  = threadIdx.x;
  const int lane = tid & 31;
  const int wid  = tid >> 5;                 // 0..15 waves per WG
  const int gw   = blockIdx.x * 16 + wid;    // 0..255 global wave id
  const int ln15 = lane & 15;                // WMMA M / N lane index
  const int koff = (lane >> 4) * 2;          // WMMA K sub-offset per ISA layout

  const v8f V8Z = {0.f,0.f,0.f,0.f,0.f,0.f,0.f,0.f};

  for (int t = 0; t < TSTEPS; ++t) {
    // ================= Stage A: gates = [x_t|last_read]@Wih^T + h@Whh^T + b =================
    {
      const int tile  = gw >> 1;
      const int half  = gw & 1;
      const int nbase = tile * 16;
      v8f acc = V8Z;
      if (half == 0) {
        const float* aP = X + ((size_t)ln15 * TSTEPS + t) * INDIM + koff;
        const float* bP = Wih + (size_t)(nbase + ln15) * HID + koff;
        acc = wmma_loop(aP, bP, INDIM, acc);                         // x_t part (k 0..255)
        const float* aP2 = CAT + ln15 * CATW + HID + koff;           // last_read part
        acc = wmma_loop(aP2, bP + INDIM, INDIM, acc);
      } else {
        const float* aP = Hs + ln15 * HID + koff;
        const float* bP = Whh + (size_t)(nbase + ln15) * HID + koff;
        acc = wmma_loop(aP, bP, HID, acc);
      }
#pragma unroll
      for (int v = 0; v < 8; ++v) sm[wid * 256 + v * 32 + lane] = acc[v];
      __syncthreads();
      if ((wid & 1) == 0) {
        const int col = nbase + ln15;
        const float bias = bih[col] + bhh[col];
#pragma unroll
        for (int v = 0; v < 8; ++v) {
          const int row = v + ((lane >> 4) << 3);
          const float val = sm[wid * 256 + v * 32 + lane] + sm[(wid + 1) * 256 + v * 32 + lane];
          GATES[row * G4H + col] = val + bias;
        }
      }
    }
    grid_sync(cnt, gen);

    // ================= Stage B: LSTM pointwise =================
    {
      const int id = blockIdx.x * NTHR + tid;     // 0..8191 == BATCH*HID
      const int bb = id >> 9, nn = id & 511;
      const float* gb = GATES + bb * G4H;
      float cc = Cs[id];
      const float gi_ = gb[nn], gf_ = gb[512 + nn], gg_ = gb[1024 + nn], go_ = gb[1536 + nn];
      cc = sigmoidf_(gf_) * cc + sigmoidf_(gi_) * tanhf(gg_);
      Cs[id] = cc;
      Hs[id] = sigmoidf_(go_) * tanhf(cc);
    }
    grid_sync(cnt, gen);

    // ================= Stage C: out = h@Wout^T + b ; xi = h@Wint^T + b =================
    {
      const int task = gw;
      const bool active = task < 124;
      const bool isOut  = task < 64;
      int nbase = 0, half = 0;
      v8f acc = V8Z;
      if (active) {
        if (isOut) { nbase = (task >> 1) * 16; half = task & 1; }
        else       { const int tt = task - 64; nbase = (tt >> 1) * 16; half = tt & 1; }
        const float* aP = Hs + ln15 * HID + half * 256 + koff;
        const float* bP;
        if (isOut) bP = Wout + (size_t)(nbase + ln15) * HID + half * 256 + koff;
        else {
          int wr = nbase + ln15; if (wr > IFACE - 1) wr = IFACE - 1;   // clamp OOB rows
          bP = Wint + (size_t)wr * HID + half * 256 + koff;
        }
        acc = wmma_loop(aP, bP, 256, acc);
      }
#pragma unroll
      for (int v = 0; v < 8; ++v) sm[wid * 256 + v * 32 + lane] = acc[v];
      __syncthreads();
      if (active && ((wid & 1) == 0)) {
        const int col = nbase + ln15;
#pragma unroll
        for (int v = 0; v < 8; ++v) {
          const int row = v + ((lane >> 4) << 3);
          const float val = sm[wid * 256 + v * 32 + lane] + sm[(wid + 1) * 256 + v * 32 + lane];
          if (isOut) CAT[row * CATW + col] = val + bout[col];
          else if (col < IFACE) XI[row * IFP + col] = val + bint[col];
        }
      }
    }
    grid_sync(cnt, gen);

    // ================= Stage D: memory module, one WG per batch element =================
    {
      const int b = blockIdx.x;
      const int j = tid;                       // memory slot / generic index (512)
      float* sIfc  = sm + S_IFACE;
      float* sRW   = sm + S_RW;
      float* sWWb  = sm + S_WW;
      float* sPrec = sm + S_PREC;
      float* sWcw  = sm + S_WCW;
      float* sAll  = sm + S_ALLOC;
      float* sFwd  = sm + S_FWD;
      float* sBwd  = sm + S_BWD;
      float* sCw   = sm + S_CW;
      float* sKey  = sm + S_SORT;
      float* sScan = sm + S_SCAN;
      float* sRed  = sm + S_RED;
      float* sMisc = sm + S_MISC;

      // --- D1: interface activations ---
      if (tid < IFACE) {
        const float v = XI[b * IFP + tid];
        float tv;
        if      (tid < 256) tv = tanhf(v);        // read keys
        else if (tid < 260) tv = softplusf_(v);   // read strengths
        else if (tid < 324) tv = tanhf(v);        // write key
        else if (tid == 324) tv = softplusf_(v);  // write strength
        else if (tid < 389) tv = sigmoidf_(v);    // erase
        else if (tid < 453) tv = tanhf(v);        // write vector
        else if (tid < 459) tv = sigmoidf_(v);    // free gates, alloc gate, write gate
        else                tv = v;               // read modes (softmax below)
        sIfc[tid] = tv;
      }
      __syncthreads();
      if (tid < 4) {                              // read-mode softmax per head
        const float a0 = sIfc[459 + tid * 3 + 0];
        const float a1 = sIfc[459 + tid * 3 + 1];
        const float a2 = sIfc[459 + tid * 3 + 2];
        const float mx = fmaxf(a0, fmaxf(a1, a2));
        const float e0 = expf(a0 - mx), e1 = expf(a1 - mx), e2 = expf(a2 - mx);
        const float s  = e0 + e1 + e2;
        sMisc[MISC_MODES + tid * 3 + 0] = e0 / s;
        sMisc[MISC_MODES + tid * 3 + 1] = e1 / s;
        sMisc[MISC_MODES + tid * 3 + 2] = e2 / s;
      }
      if (tid < 320) {                            // 5 key norms (4 read + write)
        const int seg = tid >> 6, kk = tid & 63;
        const float kv = (seg < 4) ? sIfc[seg * 64 + kk] : sIfc[260 + kk];
        sRed[tid] = kv * kv;
      }
      __syncthreads();
      for (int s = 32; s > 0; s >>= 1) {
        if (tid < 320 && (tid & 63) < s) sRed[tid] += sRed[tid + s];
        __syncthreads();
      }
      if (tid < 5) sMisc[MISC_NORM + tid] = sqrtf(sRed[tid * 64]) + DNC_DELTA;
      __syncthreads();

      // --- D2: usage update (+ stage previous read_w in LDS) ---
      float uVal;
      {
        float u = USAGE[b * MEMN + j];
        const float wwp = WW[b * MEMN + j];        // previous write weights (1 head)
        u = u + (1.f - u) * wwp;
        float psi = 1.f;
#pragma unroll
        for (int r = 0; r < RH; ++r) {
          const float rw = RW[((size_t)b * RH + r) * MEMN + j];
          sRW[r * MEMN + j] = rw;
          psi *= (1.f - sIfc[453 + r] * rw);
        }
        u *= psi;
        USAGE[b * MEMN + j] = u;
        uVal = u;
      }
      __syncthreads();

      // --- D3: write content weights (cosine + softmax over 512 slots) ---
      float wscore;
      {
        const float* mrow = MEMB + ((size_t)b * MEMN + j) * CELLW;
        float d = 0.f, m2 = 0.f;
#pragma unroll 8
        for (int k = 0; k < CELLW; ++k) { const float mv = mrow[k]; d += mv * sIfc[260 + k]; m2 += mv * mv; }
        const float mn = sqrtf(m2) + DNC_DELTA;
        wscore = (d / (mn * sMisc[MISC_NORM + 4])) * sIfc[324];
      }
      sRed[j] = wscore; __syncthreads();
      for (int s = 256; s > 0; s >>= 1) { if (j < s) sRed[j] = fmaxf(sRed[j], sRed[j + s]); __syncthreads(); }
      const float wmx = sRed[0]; __syncthreads();
      const float we = expf(wscore - wmx);
      sRed[j] = we; __syncthreads();
      for (int s = 256; s > 0; s >>= 1) { if (j < s) sRed[j] += sRed[j + s]; __syncthreads(); }
      const float wsum = sRed[0]; __syncthreads();
      sWcw[j] = we / wsum;

      // --- D4: allocation weights (bitonic argsort + exclusive product scan) ---
      sKey[j] = DNC_DELTA + (1.f - DNC_DELTA) * uVal;
      sIdx[j] = j;
      __syncthreads();
      for (int k = 2; k <= MEMN; k <<= 1) {
        for (int jj = k >> 1; jj > 0; jj >>= 1) {
          const int p = j ^ jj;
          if (p > j) {
            const bool dirUp = ((j & k) == 0);
            const float ka = sKey[j], kb = sKey[p];
            const int ia = sIdx[j], ib = sIdx[p];
            const bool gt = (ka > kb) || (ka == kb && ia > ib);   // stable tiebreak
            if (gt == dirUp) { sKey[j] = kb; sKey[p] = ka; sIdx[j] = ib; sIdx[p] = ia; }
          }
          __syncthreads();
        }
      }
      sScan[j] = sKey[j]; __syncthreads();
      for (int s = 1; s < MEMN; s <<= 1) {            // Hillis-Steele product scan
        const float tv = (j >= s) ? sScan[j - s] : 1.f;
        __syncthreads();
        sScan[j] *= tv;
        __syncthreads();
      }
      {
        const float excl = (j == 0) ? 1.f : sScan[j - 1];
        sAll[sIdx[j]] = (1.f - sKey[j]) * excl;        // scatter == inverse-perm gather
      }
      __syncthreads();

      // --- D5: new write weights + sum reduction ---
      {
        const float ag = sIfc[457], wgate = sIfc[458];
        const float wwj = wgate * (ag * sAll[j] + (1.f - ag) * sWcw[j]);
        sWWb[j] = wwj;
        WW[b * MEMN + j] = wwj;
        sRed[j] = wwj;
      }
      __syncthreads();
      for (int s = 256; s > 0; s >>= 1) { if (j < s) sRed[j] += sRed[j + s]; __syncthreads(); }
      if (j == 0) sMisc[MISC_SUMWW] = sRed[0];
      __syncthreads();

      // --- D6: memory write (erase + add), load old precedence ---
      {
        float* mrow = MEMB + ((size_t)b * MEMN + j) * CELLW;
        const float wwv = sWWb[j];
#pragma unroll 8
        for (int k = 0; k < CELLW; ++k)
          mrow[k] = mrow[k] * (1.f - wwv * sIfc[325 + k]) + wwv * sIfc[389 + k];
      }
      sPrec[j] = PRECB[b * MEMN + j];
      __threadfence_block();
      __syncthreads();

      // --- D7: link matrix update fused with fwd = L_new @ read_w_prev^T ---
      {
        const float wwi = sWWb[j];
        float f0 = 0.f, f1 = 0.f, f2 = 0.f, f3 = 0.f;
        float* Lrow = LINKB + ((size_t)b * MEMN + j) * MEMN;
        for (int c4 = 0; c4 < MEMN; c4 += 4) {
          float4 lv = *reinterpret_cast<const float4*>(Lrow + c4);
          float le[4] = {lv.x, lv.y, lv.z, lv.w};
#pragma unroll
          for (int q = 0; q < 4; ++q) {
            const int cc = c4 + q;
            float nl = (1.f - wwi - sWWb[cc]) * le[q] + wwi * sPrec[cc];
            if (cc == j) nl = 0.f;
            le[q] = nl;
            f0 += nl * sRW[0 * MEMN + cc];
            f1 += nl * sRW[1 * MEMN + cc];
            f2 += nl * sRW[2 * MEMN + cc];
            f3 += nl * sRW[3 * MEMN + cc];
          }
          lv.x = le[0]; lv.y = le[1]; lv.z = le[2]; lv.w = le[3];
          *reinterpret_cast<float4*>(Lrow + c4) = lv;
        }
        sFwd[0 * MEMN + j] = f0; sFwd[1 * MEMN + j] = f1;
        sFwd[2 * MEMN + j] = f2; sFwd[3 * MEMN + j] = f3;
      }
      __threadfence_block();
      __syncthreads();

      // --- D8: bwd = read_w_prev @ L_new (column walk, coalesced) ; D9: precedence ---
      {
        float b0 = 0.f, b1 = 0.f, b2 = 0.f, b3 = 0.f;
        for (int i = 0; i < MEMN; ++i) {
          const float lv = LINKB[((size_t)b * MEMN + i) * MEMN + j];
          b0 += sRW[0 * MEMN + i] * lv; b1 += sRW[1 * MEMN + i] * lv;
          b2 += sRW[2 * MEMN + i] * lv; b3 += sRW[3 * MEMN + i] * lv;
        }
        sBwd[0 * MEMN + j] = b0; sBwd[1 * MEMN + j] = b1;
        sBwd[2 * MEMN + j] = b2; sBwd[3 * MEMN + j] = b3;
      }
      PRECB[b * MEMN + j] = (1.f - sMisc[MISC_SUMWW]) * sPrec[j] + sWWb[j];
      __syncthreads();

      // --- D10: read content weights over NEW memory (4 heads) ---
      {
        const float* mrow = MEMB + ((size_t)b * MEMN + j) * CELLW;
        float d0 = 0.f, d1 = 0.f, d2 = 0.f, d3 = 0.f, m2 = 0.f;
#pragma unroll 8
        for (int k = 0; k < CELLW; ++k) {
          const float mv = mrow[k];
          d0 += mv * sIfc[0 * 64 + k]; d1 += mv * sIfc[1 * 64 + k];
          d2 += mv * sIfc[2 * 64 + k]; d3 += mv * sIfc[3 * 64 + k];
          m2 += mv * mv;
        }
        const float mn = sqrtf(m2) + DNC_DELTA;
        float sc[4];
        sc[0] = d0 / (mn * sMisc[MISC_NORM + 0]) * sIfc[256 + 0];
        sc[1] = d1 / (mn * sMisc[MISC_NORM + 1]) * sIfc[256 + 1];
        sc[2] = d2 / (mn * sMisc[MISC_NORM + 2]) * sIfc[256 + 2];
        sc[3] = d3 / (mn * sMisc[MISC_NORM + 3]) * sIfc[256 + 3];
#pragma unroll
        for (int r = 0; r < RH; ++r) {
          sRed[j] = sc[r]; __syncthreads();
          for (int s = 256; s > 0; s >>= 1) { if (j < s) sRed[j] = fmaxf(sRed[j], sRed[j + s]); __syncthreads(); }
          const float mx = sRed[0]; __syncthreads();
          const float e = expf(sc[r] - mx);
          sRed[j] = e; __syncthreads();
          for (int s = 256; s > 0; s >>= 1) { if (j < s) sRed[j] += sRed[j + s]; __syncthreads(); }
          const float su = sRed[0]; __syncthreads();
          sCw[r * MEMN + j] = e / su;
        }
      }
      __syncthreads();

      // --- D11: new read weights ---
#pragma unroll
      for (int r = 0; r < RH; ++r) {
        const float rwn = sMisc[MISC_MODES + r * 3 + 0] * sBwd[r * MEMN + j]
                        + sMisc[MISC_MODES + r * 3 + 1] * sFwd[r * MEMN + j]
                        + sMisc[MISC_MODES + r * 3 + 2] * sCw[r * MEMN + j];
        RW[((size_t)b * RH + r) * MEMN + j] = rwn;
        sRW[r * MEMN + j] = rwn;                 // reuse staging as NEW read_w
      }
      __syncthreads();

      // --- D12: read vectors -> CAT[:,512:768] (== last_read for next step) ---
      if (tid < RH * CELLW) {
        const int r = tid >> 6, k = tid & 63;
        float acc = 0.f;
        for (int jj = 0; jj < MEMN; ++jj)
          acc += sRW[r * MEMN + jj] * MEMB[((size_t)b * MEMN + jj) * CELLW + k];
        CAT[b * CATW + HID + r * CELLW + k] = acc;
      }
    }
    grid_sync(cnt, gen);

    // ================= Stage E: y = [out|read_vecs] @ Wmem^T + b -> d_out =================
    {
      const int task = gw;
      const bool active = task < 32;
      int nbase = 0, half = 0;
      v8f acc = V8Z;
      if (active) {
        nbase = (task >> 1) * 16; half = task & 1;
        const float* aP = CAT + ln15 * CATW + half * 384 + koff;
        const float* bP = Wmem + (size_t)(nbase + ln15) * CATW + half * 384 + koff;
        acc = wmma_loop(aP, bP, 384, acc);
      }
#pragma unroll
      for (int v = 0; v < 8; ++v) sm[wid * 256 + v * 32 + lane] = acc[v];
      __syncthreads();
      if (active && ((wid & 1) == 0)) {
        const int col = nbase + ln15;
#pragma unroll
        for (int v = 0; v < 8; ++v) {
          const int row = v + ((lane >> 4) << 3);
          const float val = sm[wid * 256 + v * 32 + lane] + sm[(wid + 1) * 256 + v * 32 + lane];
          OUTP[((size_t)row * TSTEPS + t) * INDIM + col] = val + bmem[col];
        }
      }
    }
    grid_sync(cnt, gen);
  }
}

extern "C" void kernel_launch(void* const* d_in, const int* in_sizes, int n_in,
                              void* d_out, int out_size, void* d_ws, size_t ws_size,
                              hipStream_t stream) {
  (void)in_sizes; (void)n_in; (void)out_size;
  const float* X    = (const float*)d_in[0];
  const float* Wih  = (const float*)d_in[1];
  const float* bih  = (const float*)d_in[2];
  const float* Whh  = (const float*)d_in[3];
  const float* bhh  = (const float*)d_in[4];
  const float* Wout = (const float*)d_in[5];
  const float* bout = (const float*)d_in[6];
  const float* Wint = (const float*)d_in[7];
  const float* bint = (const float*)d_in[8];
  const float* Wmem = (const float*)d_in[9];
  const float* bmem = (const float*)d_in[10];

  // zero recurrent state + barrier counters every call (deterministic, capture-safe)
  size_t need = WS_TOTALF * sizeof(float);
  if (need > ws_size) need = ws_size;
  hipMemsetAsync(d_ws, 0, need, stream);

  dnc_persistent<<<dim3(NWG), dim3(NTHR), 0, stream>>>(
      X, Wih, bih, Whh, bhh, Wout, bout, Wint, bint, Wmem, bmem,
      (float*)d_out, (float*)d_ws);
}